// AttnDecoder_1494648619343
// MI455X (gfx1250) — compile-verified
//
#include <hip/hip_runtime.h>
#include <math.h>

#define Hd 1024
#define Ed 512
#define Ad 512
#define Ld 512
#define Vd 50257
#define CAT1 (Ed + Hd)          /* 1536 */
#define COMB (Ed + Hd + Ad)     /* 2048 */

typedef __attribute__((ext_vector_type(2))) float v2f;
typedef __attribute__((ext_vector_type(8))) float v8f;

// ---- ws float layout ----------------------------------------------------
// [0,    2048)  comb vector:  [embedded(512) | attn_applied(1024) | av_emb(512)]
// [2048, 2560)  attn_pre (512)
// [2560, 3072)  attn_weights (512)
// [3072, 7168)  gates (4096)
// [7168, 8192)  x = relu(comb_W@comb + b)  (1024)
// [8192, 9216)  h_new (1024)
// [9216,10240)  c_new (1024)
#define WS_COMB   0
#define WS_PRE    2048
#define WS_ATTNW  2560
#define WS_GATES  3072
#define WS_X      7168
#define WS_HNEW   8192
#define WS_CNEW   9216

// ---- fp32 WMMA GEMV core ------------------------------------------------
// One wave computes 16 rows [row0, row0+16) of W[M x K] @ v[K].
// A tile layout (V_WMMA_F32_16X16X4_F32): lanes 0-15: M=lane, K={0,1};
// lanes 16-31: M=lane-16, K={2,3}. B is v replicated across all 16 columns,
// so per-lane B indexing matches A's K indexing and every D column holds the
// same 16 dot products.
__device__ __forceinline__ v8f wmma_gemv_acc(const float* __restrict__ W,
                                             int ldw, int K,
                                             const float* __restrict__ vvec,
                                             int row0, int M, v8f c) {
  const int lane  = threadIdx.x & 31;
  const int m     = lane & 15;
  const int khalf = lane >> 4;            // 0 -> K{0,1}, 1 -> K{2,3}
  int row = row0 + m;
  if (row > M - 1) row = M - 1;           // clamp: OOB rows discarded at store
  const float* wr = W + (size_t)row * ldw + 2 * khalf;
  const float* vr = vvec + 2 * khalf;

  v8f c2 = {};                            // second accumulator (D is plain f32
                                          // per lane, so elementwise add is OK)
  for (int k0 = 0; k0 < K; k0 += 8) {
    v2f a0, b0, a1, b1;
    a0.x = wr[k0];     a0.y = wr[k0 + 1];
    b0.x = vr[k0];     b0.y = vr[k0 + 1];
    a1.x = wr[k0 + 4]; a1.y = wr[k0 + 5];
    b1.x = vr[k0 + 4]; b1.y = vr[k0 + 5];
    c  = __builtin_amdgcn_wmma_f32_16x16x4_f32(false, a0, false, b0,
                                               (short)0, c,  false, false);
    c2 = __builtin_amdgcn_wmma_f32_16x16x4_f32(false, a1, false, b1,
                                               (short)0, c2, false, false);
  }
#pragma unroll
  for (int i = 0; i < 8; ++i) c[i] += c2[i];
  return c;
}

// D layout: VGPR i, lanes 0-15 -> M=i; lanes 16-31 -> M=8+i (all columns
// identical because B was replicated). Lane 0 stores rows 0-7, lane 16 rows 8-15.
__device__ __forceinline__ void wmma_gemv_store(v8f c,
                                                const float* __restrict__ bias,
                                                const float* __restrict__ bias2,
                                                float* __restrict__ out,
                                                int row0, int M, bool relu) {
  const int lane  = threadIdx.x & 31;
  const int khalf = lane >> 4;
  if ((lane & 15) == 0) {
#pragma unroll
    for (int i = 0; i < 8; ++i) {
      int r = row0 + khalf * 8 + i;
      if (r < M) {
        float val = c[i];
        if (bias)  val += bias[r];
        if (bias2) val += bias2[r];
        if (relu)  val = fmaxf(val, 0.0f);
        out[r] = val;
      }
    }
  }
}

// ---- K0: embedding gather + av_emb placement ----------------------------
__global__ void k_embed(const int* __restrict__ word_p,
                        const float* __restrict__ emb_table,
                        const float* __restrict__ av_emb,
                        float* __restrict__ ws) {
  const int t = threadIdx.x;                  // 512 threads
  const int word = word_p[0];
  ws[WS_COMB + t]               = emb_table[(size_t)word * Ed + t];
  ws[WS_COMB + Ed + Hd + t]     = av_emb[t];
}

// ---- K1: attn_pre[l] = dot(attn_W[l], [embedded; h0]) + attn_b[l] -------
__global__ void k_attn_pre(const int* __restrict__ word_p,
                           const float* __restrict__ emb_table,
                           const float* __restrict__ h0,
                           const float* __restrict__ attn_W,
                           const float* __restrict__ attn_b,
                           float* __restrict__ ws) {
  const int wave = (blockIdx.x * blockDim.x + threadIdx.x) >> 5; // 512 waves
  const int lane = threadIdx.x & 31;
  if (wave >= Ld) return;
  const int word = word_p[0];
  const float* wr = attn_W + (size_t)wave * CAT1;
  float acc = 0.0f;
  for (int k = lane; k < CAT1; k += 32) {
    float xv = (k < Ed) ? emb_table[(size_t)word * Ed + k] : h0[k - Ed];
    acc += wr[k] * xv;
  }
  for (int off = 16; off > 0; off >>= 1) acc += __shfl_xor(acc, off, 32);
  if (lane == 0) ws[WS_PRE + wave] = acc + attn_b[wave];
}

// ---- K2: softmax over 512 scores (single block, 16 waves) ---------------
__global__ void k_softmax(float* __restrict__ ws, float* __restrict__ out) {
  __shared__ float smax[16];
  __shared__ float ssum[16];
  __shared__ float bcast[2];
  const int t = threadIdx.x;                  // 512 threads
  float v = ws[WS_PRE + t];

  float m = v;
  for (int off = 16; off > 0; off >>= 1) m = fmaxf(m, __shfl_xor(m, off, 32));
  if ((t & 31) == 0) smax[t >> 5] = m;
  __syncthreads();
  if (t == 0) {
    float mm = smax[0];
    for (int i = 1; i < 16; ++i) mm = fmaxf(mm, smax[i]);
    bcast[0] = mm;
  }
  __syncthreads();

  float e = expf(v - bcast[0]);
  float s = e;
  for (int off = 16; off > 0; off >>= 1) s += __shfl_xor(s, off, 32);
  if ((t & 31) == 0) ssum[t >> 5] = s;
  __syncthreads();
  if (t == 0) {
    float tot = 0.0f;
    for (int i = 0; i < 16; ++i) tot += ssum[i];
    bcast[1] = tot;
  }
  __syncthreads();

  float w = e / bcast[1];
  ws[WS_ATTNW + t] = w;
  out[Vd + 2 * Hd + t] = w;                   // attn_weights output slot
}

// ---- K3: attn_applied[j] = sum_l w[l] * enc[l][j] -> comb[Ed + j] -------
__global__ void k_attn_applied(const float* __restrict__ enc,
                               float* __restrict__ ws) {
  const int j = blockIdx.x * blockDim.x + threadIdx.x;   // 1024 outputs
  if (j >= Hd) return;
  const float* w = ws + WS_ATTNW;
  float acc = 0.0f;
  for (int l = 0; l < Ld; ++l) acc += w[l] * enc[(size_t)l * Hd + j];
  ws[WS_COMB + Ed + j] = acc;
}

// ---- K4: x = relu(comb_W @ comb + comb_b)  (WMMA fp32 GEMV) -------------
__global__ void k_comb(const float* __restrict__ comb_W,
                       const float* __restrict__ comb_b,
                       float* __restrict__ ws) {
  const int wave = (blockIdx.x * blockDim.x + threadIdx.x) >> 5;
  const int row0 = wave * 16;
  if (row0 >= Hd) return;
  v8f c = {};
  c = wmma_gemv_acc(comb_W, COMB, COMB, ws + WS_COMB, row0, Hd, c);
  wmma_gemv_store(c, comb_b, nullptr, ws + WS_X, row0, Hd, true);
}

// ---- K5a: gates = W_ih@x + W_hh@h0 + b_ih + b_hh  (WMMA fp32 GEMV) ------
__global__ void k_gates(const float* __restrict__ W_ih,
                        const float* __restrict__ W_hh,
                        const float* __restrict__ b_ih,
                        const float* __restrict__ b_hh,
                        const float* __restrict__ h0,
                        float* __restrict__ ws) {
  const int wave = (blockIdx.x * blockDim.x + threadIdx.x) >> 5;
  const int row0 = wave * 16;
  if (row0 >= 4 * Hd) return;
  v8f c = {};
  c = wmma_gemv_acc(W_ih, Hd, Hd, ws + WS_X, row0, 4 * Hd, c);
  c = wmma_gemv_acc(W_hh, Hd, Hd, h0,       row0, 4 * Hd, c);
  wmma_gemv_store(c, b_ih, b_hh, ws + WS_GATES, row0, 4 * Hd, false);
}

// ---- K5b: LSTM pointwise (torch gate order i,f,g,o) ---------------------
__global__ void k_lstm(const float* __restrict__ c0,
                       float* __restrict__ ws, float* __restrict__ out) {
  const int i = blockIdx.x * blockDim.x + threadIdx.x;   // 1024
  if (i >= Hd) return;
  const float* g = ws + WS_GATES;
  float gi = g[i], gf = g[Hd + i], gg = g[2 * Hd + i], go = g[3 * Hd + i];
  float si = 1.0f / (1.0f + expf(-gi));
  float sf = 1.0f / (1.0f + expf(-gf));
  float so = 1.0f / (1.0f + expf(-go));
  float cn = sf * c0[i] + si * tanhf(gg);
  float hn = so * tanhf(cn);
  ws[WS_HNEW + i] = hn;
  ws[WS_CNEW + i] = cn;
  out[Vd + i]      = hn;                      // h_new output slot
  out[Vd + Hd + i] = cn;                      // c_new output slot
}

// ---- K6: logits = out_W @ h_new + out_b  (82% of all bytes) -------------
__global__ void k_logits(const float* __restrict__ out_W,
                         const float* __restrict__ out_b,
                         const float* __restrict__ ws,
                         float* __restrict__ out) {
  const int wave = (blockIdx.x * blockDim.x + threadIdx.x) >> 5;
  const int row0 = wave * 16;
  if (row0 >= Vd) return;                     // wave-uniform; EXEC full at WMMA
  v8f c = {};
  c = wmma_gemv_acc(out_W, Hd, Hd, ws + WS_HNEW, row0, Vd, c);
  wmma_gemv_store(c, out_b, nullptr, out, row0, Vd, false);
}

// -------------------------------------------------------------------------
extern "C" void kernel_launch(void* const* d_in, const int* in_sizes, int n_in,
                              void* d_out, int out_size, void* d_ws, size_t ws_size,
                              hipStream_t stream) {
  const int*   word      = (const int*)  d_in[0];
  // d_in[1] = enc_seq_len (==1): the row-mask in the reference is a no-op
  // for batch dim 1 with enc_seq_len >= 1, so it is intentionally unused.
  const float* av_emb    = (const float*)d_in[2];
  const float* h0        = (const float*)d_in[3];
  const float* c0        = (const float*)d_in[4];
  const float* enc       = (const float*)d_in[5];
  const float* emb_table = (const float*)d_in[6];
  const float* attn_W    = (const float*)d_in[7];
  const float* attn_b    = (const float*)d_in[8];
  const float* comb_W    = (const float*)d_in[9];
  const float* comb_b    = (const float*)d_in[10];
  const float* W_ih      = (const float*)d_in[11];
  const float* W_hh      = (const float*)d_in[12];
  const float* b_ih      = (const float*)d_in[13];
  const float* b_hh      = (const float*)d_in[14];
  const float* out_W     = (const float*)d_in[15];
  const float* out_b     = (const float*)d_in[16];
  float* ws  = (float*)d_ws;
  float* out = (float*)d_out;

  k_embed       <<<1,   512, 0, stream>>>(word, emb_table, av_emb, ws);
  k_attn_pre    <<<64,  256, 0, stream>>>(word, emb_table, h0, attn_W, attn_b, ws);
  k_softmax     <<<1,   512, 0, stream>>>(ws, out);
  k_attn_applied<<<4,   256, 0, stream>>>(enc, ws);
  k_comb        <<<8,   256, 0, stream>>>(comb_W, comb_b, ws);        // 64 tiles
  k_gates       <<<32,  256, 0, stream>>>(W_ih, W_hh, b_ih, b_hh, h0, ws); // 256 tiles
  k_lstm        <<<4,   256, 0, stream>>>(c0, ws, out);
  // logits: 3142 tiles of 16 rows, 8 waves/block -> 393 blocks
  k_logits      <<<393, 256, 0, stream>>>(out_W, out_b, ws, out);
}